// Linear_86397562126687
// MI455X (gfx1250) — compile-verified
//
#include <hip/hip_runtime.h>

// CDNA5 (gfx1250) equivariant linear layer: three block-diagonal fp32 GEMMs
//   block b: Y[(n,i),v] = alpha * sum_u X[(n,i),u] * W_b[u,v],  alpha = 1/16
// shapes: (32768*d) x 256 x 256 for d in {1,3,5}
// V_WMMA_F32_16X16X4_F32 (exact fp32) + LDS tiling; W tile stored transposed
// so both A and B fragments are single ds_load_b64 ops (no shuffle movs).

typedef __attribute__((ext_vector_type(2))) float v2f;
typedef __attribute__((ext_vector_type(4))) float v4f;
typedef __attribute__((ext_vector_type(8))) float v8f;

#define IN_DIM  2304
#define MUL     256     // mul per irrep (K and N of every GEMM)
#define TM      128     // rows per workgroup tile
#define TN      128     // cols per workgroup tile
#define TK      32      // K chunk staged in LDS
#define APAD    36      // A row stride in LDS floats: 144B -> 16B aligned, banks spread
#define BPAD    36      // B(col-major) row stride in LDS floats
#define NTHREADS 256    // 8 waves (wave32)

template <int D>
__global__ __launch_bounds__(NTHREADS)
void eqlin_wmma_f32(const float* __restrict__ x,
                    const float* __restrict__ w,
                    float* __restrict__ y,
                    int xo, int wo)
{
    __shared__ float Alds[TM * APAD];   // [row][k], padded
    __shared__ float Blds[TN * BPAD];   // [col][k] (transposed), padded

    const int tid  = threadIdx.x;
    const int lane = tid & 31;
    const int wave = tid >> 5;
    const int wr   = wave >> 1;          // 0..3 : 32-row strip
    const int wc   = wave & 1;           // 0..1 : 64-col strip

    const int r0 = blockIdx.x * TM;      // global row base (rows are n*D + i)
    const int c0 = blockIdx.y * TN;      // global col base

    const int mL = lane & 15;            // M (or col) index inside a 16-wide tile
    const int kg = (lane >> 4) << 1;     // K sub-offset per ISA f32 layout (0 or 2)

    v8f acc[2][4] = {};                  // 2 row-tiles x 4 col-tiles of 16x16 f32

    for (int k0 = 0; k0 < MUL; k0 += TK) {
        __syncthreads();                 // previous chunk fully consumed

        // ---- stage A[TM x TK] ----
        if (D == 1) {
            // contiguous K: 128-bit global loads + 128-bit LDS stores
            #pragma unroll
            for (int e = tid * 4; e < TM * TK; e += NTHREADS * 4) {
                const int rl = e >> 5;       // local row == local n
                const int k  = e & (TK - 1);
                const v4f v  = *(const v4f*)&x[(r0 + rl) * IN_DIM + xo + k0 + k];
                *(v4f*)&Alds[rl * APAD + k] = v;
            }
        } else {
            // d-strided gather; n-major rows keep unique-byte DRAM traffic
            #pragma unroll
            for (int e = tid; e < TM * TK; e += NTHREADS) {
                const int rl = e >> 5;
                const int k  = e & (TK - 1);
                const int r  = r0 + rl;
                const int n  = r / D;
                const int i  = r - n * D;
                Alds[rl * APAD + k] = x[n * IN_DIM + xo + (k0 + k) * D + i];
            }
        }

        // ---- stage W[TK x TN] transposed into [col][k] ----
        // global side coalesced (512B rows); LDS store strided (cheap)
        #pragma unroll
        for (int e = tid; e < TK * TN; e += NTHREADS) {
            const int kk = e >> 7;           // k row
            const int c  = e & (TN - 1);     // col
            Blds[c * BPAD + kk] = w[wo + (k0 + kk) * MUL + (c0 + c)];
        }
        __syncthreads();

        // ---- compute: K in steps of 4 via V_WMMA_F32_16X16X4_F32 ----
        #pragma unroll
        for (int k = 0; k < TK; k += 4) {
            v2f a[2];
            #pragma unroll
            for (int rt = 0; rt < 2; ++rt) {
                const float* pa = &Alds[(wr * 32 + rt * 16 + mL) * APAD + k + kg];
                a[rt] = *(const v2f*)pa;                 // ds_load_b64
            }
            #pragma unroll
            for (int ct = 0; ct < 4; ++ct) {
                const int col   = wc * 64 + ct * 16 + mL;
                const float* pb = &Blds[col * BPAD + k + kg];
                const v2f b     = *(const v2f*)pb;       // ds_load_b64 (no movs)
                #pragma unroll
                for (int rt = 0; rt < 2; ++rt) {
                    acc[rt][ct] = __builtin_amdgcn_wmma_f32_16x16x4_f32(
                        /*neg_a=*/false, a[rt],
                        /*neg_b=*/false, b,
                        /*c_mod=*/(short)0, acc[rt][ct],
                        /*reuse_a=*/false, /*reuse_b=*/false);
                }
            }
        }
    }

    // ---- scale by alpha = 1/sqrt(256) and store (C/D layout: VGPR j ->
    //      M = j + 8*(lane>=16), N = lane&15) ----
    const float alpha = 0.0625f;
    #pragma unroll
    for (int rt = 0; rt < 2; ++rt) {
        #pragma unroll
        for (int ct = 0; ct < 4; ++ct) {
            #pragma unroll
            for (int j = 0; j < 8; ++j) {
                const int m = j + ((lane >> 4) << 3);
                const int R = r0 + wr * 32 + rt * 16 + m;
                const int V = c0 + wc * 64 + ct * 16 + mL;
                const int n = R / D;
                const int i = R - n * D;
                y[n * IN_DIM + xo + V * D + i] = acc[rt][ct][j] * alpha;
            }
        }
    }
}

extern "C" void kernel_launch(void* const* d_in, const int* in_sizes, int n_in,
                              void* d_out, int out_size, void* d_ws, size_t ws_size,
                              hipStream_t stream)
{
    const float* x = (const float*)d_in[0];   // [32768, 2304] fp32
    const float* w = (const float*)d_in[1];   // [196608]      fp32
    float* y       = (float*)d_out;           // [32768, 2304] fp32

    // block 0: d=1, xo=0,    wo=0,       M = 32768  -> 256 row groups
    eqlin_wmma_f32<1><<<dim3(32768 / TM, MUL / TN), NTHREADS, 0, stream>>>(x, w, y, 0, 0);
    // block 1: d=3, xo=256,  wo=65536,   M = 98304  -> 768 row groups
    eqlin_wmma_f32<3><<<dim3(98304 / TM, MUL / TN), NTHREADS, 0, stream>>>(x, w, y, 256, 65536);
    // block 2: d=5, xo=1024, wo=131072,  M = 163840 -> 1280 row groups
    eqlin_wmma_f32<5><<<dim3(163840 / TM, MUL / TN), NTHREADS, 0, stream>>>(x, w, y, 1024, 131072);
}